// GSModel_82343112999211
// MI455X (gfx1250) — compile-verified
//
#include <hip/hip_runtime.h>
#include <hip/hip_bf16.h>
#include <math.h>

typedef float v2f __attribute__((ext_vector_type(2)));
typedef float v8f __attribute__((ext_vector_type(8)));

#define REC   12      // floats per gaussian record
#define MAXG  1024    // max gaussians staged in LDS (48 KB)

#define SH_C0 0.28209479177387814f
#define SH_C1 0.4886025119029199f
#define NEARP 0.01f
#define FARP  1.0e10f
#define LOWPASS 0.3f
#define AMIN (1.0f/255.0f)
#define AMAX 0.999f

// ---------------------------------------------------------------------------
// Kernel 1: per-gaussian preprocessing.
// Record layout: [cA,cB,cC, mx,my, op, r,g,b, depth, sortkey, pad]
// ---------------------------------------------------------------------------
__global__ void gs_preprocess(const float* __restrict__ means3d,
                              const float* __restrict__ quats,
                              const float* __restrict__ scales,
                              const float* __restrict__ opac,
                              const float* __restrict__ sh,
                              const float* __restrict__ c2w,
                              const float* __restrict__ Km,
                              int N, float* __restrict__ rec)
{
    int i = blockIdx.x * blockDim.x + threadIdx.x;
    if (i >= N) return;

    // view = inverse of rigid cam-to-world: Rw = Rcw^T, tw = -Rcw^T * tcw
    float Rcw[3][3] = {{c2w[0],c2w[1],c2w[2]},
                       {c2w[4],c2w[5],c2w[6]},
                       {c2w[8],c2w[9],c2w[10]}};
    float tcw[3] = {c2w[3], c2w[7], c2w[11]};
    float Rw[3][3], tw[3];
    for (int r = 0; r < 3; ++r)
        for (int c = 0; c < 3; ++c)
            Rw[r][c] = Rcw[c][r];
    for (int r = 0; r < 3; ++r)
        tw[r] = -(Rw[r][0]*tcw[0] + Rw[r][1]*tcw[1] + Rw[r][2]*tcw[2]);

    float fx = Km[0], fy = Km[4], cx = Km[2], cy = Km[5];

    float m0 = means3d[3*i+0], m1 = means3d[3*i+1], m2 = means3d[3*i+2];
    float x = Rw[0][0]*m0 + Rw[0][1]*m1 + Rw[0][2]*m2 + tw[0];
    float y = Rw[1][0]*m0 + Rw[1][1]*m1 + Rw[1][2]*m2 + tw[1];
    float z = Rw[2][0]*m0 + Rw[2][1]*m1 + Rw[2][2]*m2 + tw[2];

    bool valid = (z > NEARP) && (z < FARP);
    float zs = valid ? z : 1.0f;
    float mx = fx * x / zs + cx;
    float my = fy * y / zs + cy;

    // quaternion -> rotation
    float qw = quats[4*i+0], qx = quats[4*i+1], qy = quats[4*i+2], qz = quats[4*i+3];
    float qn = sqrtf(qw*qw + qx*qx + qy*qy + qz*qz);
    qw /= qn; qx /= qn; qy /= qn; qz /= qn;
    float Rg[3][3] = {
        {1.f-2.f*(qy*qy+qz*qz), 2.f*(qx*qy-qw*qz),     2.f*(qx*qz+qw*qy)},
        {2.f*(qx*qy+qw*qz),     1.f-2.f*(qx*qx+qz*qz), 2.f*(qy*qz-qw*qx)},
        {2.f*(qx*qz-qw*qy),     2.f*(qy*qz+qw*qx),     1.f-2.f*(qx*qx+qy*qy)}};

    float s0 = scales[3*i+0], s1 = scales[3*i+1], s2 = scales[3*i+2];
    float M[3][3];
    for (int r = 0; r < 3; ++r) { M[r][0]=Rg[r][0]*s0; M[r][1]=Rg[r][1]*s1; M[r][2]=Rg[r][2]*s2; }
    float C3[3][3];
    for (int r = 0; r < 3; ++r)
        for (int c = 0; c < 3; ++c)
            C3[r][c] = M[r][0]*M[c][0] + M[r][1]*M[c][1] + M[r][2]*M[c][2];
    float T1[3][3], CC[3][3];
    for (int r = 0; r < 3; ++r)
        for (int c = 0; c < 3; ++c)
            T1[r][c] = Rw[r][0]*C3[0][c] + Rw[r][1]*C3[1][c] + Rw[r][2]*C3[2][c];
    for (int r = 0; r < 3; ++r)
        for (int c = 0; c < 3; ++c)
            CC[r][c] = T1[r][0]*Rw[c][0] + T1[r][1]*Rw[c][1] + T1[r][2]*Rw[c][2];

    float j00 = fx/zs, j02 = -fx*x/(zs*zs);
    float j11 = fy/zs, j12 = -fy*y/(zs*zs);
    float u00 = j00*CC[0][0] + j02*CC[2][0];
    float u01 = j00*CC[0][1] + j02*CC[2][1];
    float u02 = j00*CC[0][2] + j02*CC[2][2];
    float u11 = j11*CC[1][1] + j12*CC[2][1];
    float u12 = j11*CC[1][2] + j12*CC[2][2];
    float a  = u00*j00 + u02*j02 + LOWPASS;
    float bb = u01*j11 + u02*j12;
    float c  = u11*j11 + u12*j12 + LOWPASS;
    float det = a*c - bb*bb;
    valid = valid && (det > 0.0f);
    float dets = (det > 0.0f) ? det : 1.0f;
    float conA =  c / dets;
    float conB = -bb / dets;
    float conC =  a / dets;

    // SH color (degree 1)
    float dx = m0 - tcw[0], dy = m1 - tcw[1], dz = m2 - tcw[2];
    float dn = fmaxf(sqrtf(dx*dx + dy*dy + dz*dz), 1e-8f);
    dx /= dn; dy /= dn; dz /= dn;
    float col[3];
    for (int k = 0; k < 3; ++k) {
        float v = SH_C0 * sh[i*12 + 0*3 + k]
                - SH_C1 * dy * sh[i*12 + 1*3 + k]
                + SH_C1 * dz * sh[i*12 + 2*3 + k]
                - SH_C1 * dx * sh[i*12 + 3*3 + k];
        col[k] = fmaxf(v + 0.5f, 0.0f);
    }

    float op = 1.0f / (1.0f + __expf(-opac[i]));
    op = valid ? op : 0.0f;
    float key = valid ? z : 3.0e38f;

    float* r = rec + i*REC;
    r[0] = conA; r[1] = conB; r[2] = conC;
    r[3] = mx;   r[4] = my;   r[5] = op;
    r[6] = col[0]; r[7] = col[1]; r[8] = col[2];
    r[9] = z;    r[10] = key; r[11] = 0.0f;
}

// ---------------------------------------------------------------------------
// Kernel 2: stable rank-sort by depth key (single 1024-thread block).
// ---------------------------------------------------------------------------
__global__ void gs_sort(const float* __restrict__ rec, float* __restrict__ srec, int N)
{
    __shared__ float keys[MAXG];
    int i = threadIdx.x;
    keys[i] = (i < N) ? rec[i*REC + 10] : 3.4e38f;
    __syncthreads();
    if (i < N) {
        float ki = keys[i];
        int rank = 0;
        for (int j = 0; j < N; ++j) {
            float kj = keys[j];
            rank += (kj < ki) || (kj == ki && j < i);
        }
        for (int k = 0; k < REC; ++k)
            srec[rank*REC + k] = rec[i*REC + k];
    }
}

// ---------------------------------------------------------------------------
// Kernel 3: WMMA-tiled alpha compositing.
// Each wave owns 16 pixels; each 16-gaussian batch forms a 16x16 sigma tile
// via two V_WMMA_F32_16X16X4_F32 (rank-6 bilinear form, K padded to 8).
// ---------------------------------------------------------------------------
__global__ void __launch_bounds__(256) gs_composite(const float* __restrict__ srec,
                                                    const int* __restrict__ pW,
                                                    int N, int npix,
                                                    float* __restrict__ out)
{
    __shared__ float g[MAXG * REC];   // 48 KB
    int W = *pW;
    int Ng = (N < MAXG) ? N : MAXG;

    // Stage full sorted gaussian table into LDS (float4 granularity; REC*4B=48B stride).
    const int tot4 = (MAXG * REC) / 4;
    const int n4   = (Ng * REC) / 4;
    for (int k = threadIdx.x; k < tot4; k += blockDim.x) {
        float4 v = make_float4(0.f, 0.f, 0.f, 0.f);
        if (k < n4) v = ((const float4*)srec)[k];
        ((float4*)g)[k] = v;
    }
    __syncthreads();

    int lane = threadIdx.x & 31;
    int wave = (blockIdx.x * blockDim.x + threadIdx.x) >> 5;
    int col  = lane & 15;
    bool hi  = lane >= 16;

    int base = wave * 16;
    int pix  = base + col;
    int cpix = (pix < npix) ? pix : (npix - 1);

    // Shift coordinates by the wave base pixel to tame f32 cancellation.
    float x0 = (float)(base % W) + 0.5f;
    float y0 = (float)(base / W) + 0.5f;
    float px = (float)(cpix % W) + 0.5f - x0;
    float py = (float)(cpix / W) + 0.5f - y0;

    // Pixel features f = [px^2, py^2, px*py, px, py, 1, 0, 0]  (B matrix, K x 16)
    float pf0 = px*px, pf1 = py*py, pf2 = px*py, pf3 = px, pf4 = py, pf5 = 1.0f;
    v2f bfr0, bfr1;
    bfr0.x = hi ? pf2 : pf0;  bfr0.y = hi ? pf3 : pf1;
    bfr1.x = hi ? 0.f : pf4;  bfr1.y = hi ? 0.f : pf5;

    float T = 1.0f, aR = 0.f, aG = 0.f, aB = 0.f, aD = 0.f, aA = 0.f;

    int nb = (Ng + 15) >> 4;
    for (int b = 0; b < nb; ++b) {
        // Gaussian features for gaussian (b*16 + col):
        // gfeat = [cA/2, cC/2, cB, -(cA*mx+cB*my), -(cC*my+cB*mx), (cA*mx^2+cC*my^2)/2 + cB*mx*my]
        const float* r = &g[(b*16 + col) * REC];
        float cA = r[0], cB = r[1], cC = r[2];
        float mx = r[3] - x0, my = r[4] - y0;
        float g0 = 0.5f * cA;
        float g1 = 0.5f * cC;
        float g2 = cB;
        float g3 = -(cA*mx + cB*my);
        float g4 = -(cC*my + cB*mx);
        float g5 = 0.5f*(cA*mx*mx + cC*my*my) + cB*mx*my;

        v2f afr0, afr1;
        afr0.x = hi ? g2  : g0;  afr0.y = hi ? g3  : g1;
        afr1.x = hi ? 0.f : g4;  afr1.y = hi ? 0.f : g5;

        // sigma tile: D[M=gaussian][N=pixel], two K=4 WMMAs accumulate K=8.
        v8f sig = {};
        sig = __builtin_amdgcn_wmma_f32_16x16x4_f32(false, afr0, false, bfr0,
                                                    (short)0, sig, false, false);
        sig = __builtin_amdgcn_wmma_f32_16x16x4_f32(false, afr1, false, bfr1,
                                                    (short)0, sig, false, false);

        // Sequential transmittance over this half-wave's 8 gaussians.
        int gbase = b*16 + (hi ? 8 : 0);
        float P = 1.0f, pr = 0.f, pg = 0.f, pb = 0.f, pd = 0.f, pa = 0.f;
        #pragma unroll
        for (int rr = 0; rr < 8; ++rr) {
            const float* q = &g[(gbase + rr) * REC];
            float s  = sig[rr];
            float al = fminf(q[5] * __expf(-s), AMAX);
            al = (s >= 0.0f && al >= AMIN) ? al : 0.0f;
            float w = al * P;
            P *= (1.0f - al);
            pr += w * q[6]; pg += w * q[7]; pb += w * q[8];
            pd += w * q[9]; pa += w;
        }

        // Stitch the two 8-gaussian halves (lanes l and l+16 share a pixel):
        // total = part_lo + P_lo * part_hi ;  T *= P_lo * P_hi
        float oP  = __shfl_xor(P,  16);
        float opr = __shfl_xor(pr, 16);
        float opg = __shfl_xor(pg, 16);
        float opb = __shfl_xor(pb, 16);
        float opd = __shfl_xor(pd, 16);
        float opa2= __shfl_xor(pa, 16);

        float fr = hi ? (opr + oP*pr) : (pr + P*opr);
        float fg = hi ? (opg + oP*pg) : (pg + P*opg);
        float fb = hi ? (opb + oP*pb) : (pb + P*opb);
        float fd = hi ? (opd + oP*pd) : (pd + P*opd);
        float fa = hi ? (opa2+ oP*pa) : (pa + P*opa2);
        float fP = P * oP;

        aR += T * fr; aG += T * fg; aB += T * fb;
        aD += T * fd; aA += T * fa;
        T *= fP;
    }

    if (!hi && pix < npix) {
        out[pix*4 + 0] = aR;
        out[pix*4 + 1] = aG;
        out[pix*4 + 2] = aB;
        out[pix*4 + 3] = aD / fmaxf(aA, 1e-10f);
        out[npix*4 + pix] = aA;   // render_alphas
    }
}

// ---------------------------------------------------------------------------
extern "C" void kernel_launch(void* const* d_in, const int* in_sizes, int n_in,
                              void* d_out, int out_size, void* d_ws, size_t ws_size,
                              hipStream_t stream)
{
    const float* means3d = (const float*)d_in[0];
    const float* quats   = (const float*)d_in[1];
    const float* scales  = (const float*)d_in[2];
    const float* opac    = (const float*)d_in[3];
    const float* sh      = (const float*)d_in[4];
    const float* c2w     = (const float*)d_in[5];
    const float* Km      = (const float*)d_in[6];
    const int*   pW      = (const int*)d_in[7];
    float*       out     = (float*)d_out;

    int N    = in_sizes[0] / 3;       // gaussians
    int npix = out_size / 5;          // H*W  (4 color + 1 alpha per pixel)

    float* rec  = (float*)d_ws;                  // unsorted records
    float* srec = rec + (size_t)MAXG * REC;      // sorted records

    gs_preprocess<<<(N + 255) / 256, 256, 0, stream>>>(
        means3d, quats, scales, opac, sh, c2w, Km, N, rec);

    gs_sort<<<1, MAXG, 0, stream>>>(rec, srec, N);

    int waves  = (npix + 15) / 16;
    int blocks = (waves + 7) / 8;     // 8 waves (256 threads) per block
    gs_composite<<<blocks, 256, 0, stream>>>(srec, pW, N, npix, out);
}